// Pattern_extrator_73959336837450
// MI455X (gfx1250) — compile-verified
//
#include <hip/hip_runtime.h>

typedef __attribute__((ext_vector_type(2))) float v2f;
typedef __attribute__((ext_vector_type(8))) float v8f;
typedef __attribute__((ext_vector_type(4))) unsigned int v4u;
typedef __attribute__((ext_vector_type(8))) int v8i;
typedef __attribute__((ext_vector_type(4))) int v4i;

#define LA 1536
#define LS 512
#define LL 2048
#define CC 256
#define HH 4
#define DKK 64
#define BB 4

#if defined(__has_builtin)
#if __has_builtin(__builtin_amdgcn_tensor_load_to_lds) && \
    __has_builtin(__builtin_amdgcn_s_wait_tensorcnt)
#define HAVE_TDM 1
#endif
#endif
#ifndef HAVE_TDM
#define HAVE_TDM 0
#endif

static __device__ __forceinline__ v8f wmma_f32(v2f a, v2f b, v8f c) {
  // D = A(16x4 f32) * B(4x16 f32) + C(16x16 f32)
  return __builtin_amdgcn_wmma_f32_16x16x4_f32(false, a, false, b, (short)0, c,
                                               false, false);
}

static __device__ __forceinline__ float halfmax16(float v) {
#pragma unroll
  for (int m = 1; m < 16; m <<= 1) v = fmaxf(v, __shfl_xor(v, m, 32));
  return v;
}
static __device__ __forceinline__ float halfsum16(float v) {
#pragma unroll
  for (int m = 1; m < 16; m <<= 1) v += __shfl_xor(v, m, 32);
  return v;
}

// Ensure all prior LDS reads of a buffer have completed before TDM overwrites.
static __device__ __forceinline__ void wait_ds0() {
  asm volatile("s_wait_dscnt 0x0" ::: "memory");
}

// ---------------------------------------------------------------------------
// TDM tile load: 16 rows x 64 f32, global row stride `stride_elems`, into an
// LDS [16][68] buffer.  pad_enable inserts 4 DWORDs of padding every 64
// DWORDs, reproducing the 68-float row pitch (bank-conflict avoidance) in the
// DMA itself.  One issue per wave; tracked with TENSORcnt.
// ---------------------------------------------------------------------------
static __device__ __forceinline__ void tile_load_16x64(
    float (*dst)[68], const float* __restrict__ g, unsigned int stride_elems) {
#if HAVE_TDM
  unsigned int lds_addr = (unsigned int)(size_t)(&dst[0][0]);  // addr[31:0]
  unsigned long long ga = (unsigned long long)(size_t)g;
  v4u g0;
  g0.x = 1u;                    // count=1 (valid), no gather, no restore
  g0.y = lds_addr;              // lds_addr
  g0.z = (unsigned int)ga;      // global_addr[31:0]
  g0.w = (unsigned int)((ga >> 32) & 0x01FFFFFFu) | 0x80000000u;  // +type=2
  v8i g1;
  g1[0] = 0x07520000;           // data_size=4B | pad_en | interval=64dw | pad=4dw
  g1[1] = (int)(64u << 16);     // tensor_dim0 = 64 (low16 at bits 63:48)
  g1[2] = (int)(16u << 16);     // tensor_dim0 hi=0 | tensor_dim1 = 16
  g1[3] = (int)(64u << 16);     // tensor_dim1 hi=0 | tile_dim0 = 64
  g1[4] = 16;                   // tile_dim1 = 16, tile_dim2 = 0
  g1[5] = (int)stride_elems;    // tensor_dim0_stride (data_size units), low 32
  g1[6] = 0;                    // stride hi / dim1_stride lo (unused, 2D tile)
  g1[7] = 0;
  v4i z4 = {0, 0, 0, 0};
  v8i z8 = {0, 0, 0, 0, 0, 0, 0, 0};
  __builtin_amdgcn_tensor_load_to_lds(g0, g1, z4, z4, z8, 0);
#else
  const int lane = threadIdx.x;
#pragma unroll
  for (int t = 0; t < 8; ++t) {
    int idx = lane + 32 * t;
    int row = idx >> 4, c4 = (idx & 15) << 2;
    *(float4*)&dst[row][c4] = *(const float4*)(g + (size_t)row * stride_elems + c4);
  }
#endif
}

// Wait for staged tiles.  more!=0: leave the newest in-flight pair outstanding
// (double buffering); more==0: drain.
static __device__ __forceinline__ void wait_tiles(int more) {
#if HAVE_TDM
  if (more)
    __builtin_amdgcn_s_wait_tensorcnt(2);
  else
    __builtin_amdgcn_s_wait_tensorcnt(0);
#else
  __syncthreads();
#endif
  asm volatile("" ::: "memory");
}

// ---------------------------------------------------------------------------
// Kernel 1: fused QKV projection.  y = x @ W^T + b for W in {Wq,Wk,Wv}.
// One wave computes one 16x16 tile; K staged via TDM in 16x64 chunks.
// out -> [3][B][H][L][DK].
// ---------------------------------------------------------------------------
__global__ __launch_bounds__(32) void qkv_proj_kernel(
    const float* __restrict__ audio, const float* __restrict__ text,
    const float* __restrict__ Wq, const float* __restrict__ bq,
    const float* __restrict__ Wk, const float* __restrict__ bk,
    const float* __restrict__ Wv, const float* __restrict__ bv,
    float* __restrict__ qkv) {
  __shared__ float As[16][68];
  __shared__ float Bs[16][68];
  const int lane = threadIdx.x;
  const int lo = lane & 15, hi = lane >> 4;
  const int mbase = blockIdx.x * 16;
  const int nbase = blockIdx.y * 16;
  const int w = blockIdx.z;
  const float* W = (w == 0) ? Wq : (w == 1) ? Wk : Wv;
  const float* bias = (w == 0) ? bq : (w == 1) ? bk : bv;

  // 16-row tile never straddles batch or audio/text boundary (both 16-aligned)
  const int bidx = mbase >> 11;
  const int l0 = mbase & (LL - 1);
  const float* xbase = (l0 < LA)
                           ? (audio + ((size_t)bidx * LA + l0) * CC)
                           : (text + ((size_t)bidx * LS + (l0 - LA)) * CC);

  v8f acc = {};
  for (int kc = 0; kc < CC; kc += 64) {
    wait_ds0();  // previous chunk's LDS reads done before DMA overwrites
    tile_load_16x64(As, xbase + kc, CC);
    tile_load_16x64(Bs, W + (size_t)nbase * CC + kc, CC);
    wait_tiles(0);
#pragma unroll
    for (int kk = 0; kk < 64; kk += 4) {
      v2f a = *(const v2f*)&As[lo][kk + 2 * hi];
      v2f b = *(const v2f*)&Bs[lo][kk + 2 * hi];
      acc = wmma_f32(a, b, acc);
    }
  }

  const float bval = bias[nbase + lo];
  const size_t BHLD = (size_t)BB * HH * LL * DKK;
  const int cg = nbase + lo;
  const int h = cg >> 6;
  const int d = cg & 63;
#pragma unroll
  for (int p = 0; p < 8; ++p) {
    int mg = mbase + p + 8 * hi;
    int l = mg & (LL - 1);
    size_t off =
        (size_t)w * BHLD + (((size_t)(mg >> 11) * HH + h) * LL + l) * DKK + d;
    qkv[off] = acc[p] + bval;
  }
}

// ---------------------------------------------------------------------------
// Kernel 2: masked attention, flash-style streaming softmax over key tiles.
// K/V tiles double-buffered through the TDM.  Inverted-mask semantics of the
// reference: eq0 positions -> -inf before softmax -> 0 after.
// ---------------------------------------------------------------------------
__global__ __launch_bounds__(32) void attn_kernel(const float* __restrict__ qkv,
                                                  const int* __restrict__ ilens,
                                                  const int* __restrict__ ylens,
                                                  float* __restrict__ ctx) {
  __shared__ float Qs[16][68];
  __shared__ float Ks[2][16][68];
  __shared__ float Vs[2][16][68];
  __shared__ float Ps[16][20];
  const int lane = threadIdx.x, lo = lane & 15, hi = lane >> 4;
  const int qbase = blockIdx.x * 16;
  const int h = blockIdx.y;
  const int b = blockIdx.z;
  const int ilen = ilens[b];
  const int ylen = ylens[b];
  const size_t BHLD = (size_t)BB * HH * LL * DKK;
  const float* Qp = qkv + ((size_t)b * HH + h) * LL * DKK;
  const float* Kp = Qp + BHLD;
  const float* Vp = Qp + 2 * BHLD;

  // prologue: Q tile + first K/V pair in flight
  tile_load_16x64(Qs, Qp + (size_t)qbase * DKK, DKK);
  tile_load_16x64(Ks[0], Kp, DKK);
  tile_load_16x64(Vs[0], Vp, DKK);

  v8f o0 = {}, o1 = {}, o2 = {}, o3 = {};
  float mi[8], li[8];
#pragma unroll
  for (int p = 0; p < 8; ++p) {
    mi[p] = -__builtin_inff();
    li[p] = 0.f;
  }

  for (int j = 0; j < LL / 16; ++j) {
    const int cur = j & 1;
    const int more = (j + 1) < (LL / 16);
    if (more) {
      wait_ds0();  // buffer cur^1 was last read in iteration j-1
      tile_load_16x64(Ks[cur ^ 1], Kp + (size_t)(j + 1) * 16 * DKK, DKK);
      tile_load_16x64(Vs[cur ^ 1], Vp + (size_t)(j + 1) * 16 * DKK, DKK);
    }
    wait_tiles(more);  // tiles for j (and Q) resident; next pair may be in flight

    // S = Q K^T : both operands row-major [16][DK] -> identical fragment form
    v8f s = {};
#pragma unroll
    for (int kk = 0; kk < 64; kk += 4) {
      v2f a = *(const v2f*)&Qs[lo][kk + 2 * hi];
      v2f bb = *(const v2f*)&Ks[cur][lo][kk + 2 * hi];
      s = wmma_f32(a, bb, s);
    }

    const int jg = j * 16 + lo;  // key column held by this lane
#pragma unroll
    for (int p = 0; p < 8; ++p) {
      int ig = qbase + p + 8 * hi;
      float sv = s[p] * 0.125f;  // 1/sqrt(64)
      bool eq0;
      if (jg < LA)
        eq0 = (jg < ilen);
      else if (ig < LA)
        eq0 = false;
      else
        eq0 = ((ig - LA) >= (jg - LA)) && ((jg - LA) < ylen);
      s[p] = eq0 ? -__builtin_inff() : sv;
    }

    // online softmax update
#pragma unroll
    for (int p = 0; p < 8; ++p) {
      float rmax = halfmax16(s[p]);
      float mnew = fmaxf(mi[p], rmax);
      float muse = fmaxf(mnew, -1.0e30f);  // guard all-masked-so-far rows
      float a = __expf(mi[p] - muse);
      float pv = __expf(s[p] - muse);
      float rsum = halfsum16(pv);
      li[p] = a * li[p] + rsum;
      mi[p] = mnew;
      Ps[p + 8 * hi][lo] = pv;  // bounce P through LDS to A-fragment layout
      o0[p] *= a;
      o1[p] *= a;
      o2[p] *= a;
      o3[p] *= a;
    }
    __syncthreads();

    // O += P @ V : contraction over 16 keys -> 4 k-steps, 4 DK sub-tiles
#pragma unroll
    for (int kk = 0; kk < 16; kk += 4) {
      v2f a = *(const v2f*)&Ps[lo][kk + 2 * hi];
      int r0 = kk + 2 * hi;
      v2f b0 = {Vs[cur][r0][lo], Vs[cur][r0 + 1][lo]};
      v2f b1 = {Vs[cur][r0][16 + lo], Vs[cur][r0 + 1][16 + lo]};
      v2f b2 = {Vs[cur][r0][32 + lo], Vs[cur][r0 + 1][32 + lo]};
      v2f b3 = {Vs[cur][r0][48 + lo], Vs[cur][r0 + 1][48 + lo]};
      o0 = wmma_f32(a, b0, o0);
      o1 = wmma_f32(a, b1, o1);
      o2 = wmma_f32(a, b2, o2);
      o3 = wmma_f32(a, b3, o3);
    }
  }

  // normalize and write context [B, L, C] with heads re-interleaved
#pragma unroll
  for (int p = 0; p < 8; ++p) {
    int ig = qbase + p + 8 * hi;
    float inv = 1.0f / li[p];
    size_t base = ((size_t)b * LL + ig) * CC + (size_t)h * DKK;
    ctx[base + lo] = o0[p] * inv;
    ctx[base + 16 + lo] = o1[p] * inv;
    ctx[base + 32 + lo] = o2[p] * inv;
    ctx[base + 48 + lo] = o3[p] * inv;
  }
}

// ---------------------------------------------------------------------------
// Kernel 3: output projection  out = ctx @ Wo^T + bo   ([8192,256] x [256,256])
// ---------------------------------------------------------------------------
__global__ __launch_bounds__(32) void oproj_kernel(const float* __restrict__ ctx,
                                                   const float* __restrict__ Wo,
                                                   const float* __restrict__ bo,
                                                   float* __restrict__ out) {
  __shared__ float As[16][68];
  __shared__ float Bs[16][68];
  const int lane = threadIdx.x, lo = lane & 15, hi = lane >> 4;
  const int mbase = blockIdx.x * 16, nbase = blockIdx.y * 16;
  v8f acc = {};
  for (int kc = 0; kc < CC; kc += 64) {
    wait_ds0();
    tile_load_16x64(As, ctx + (size_t)mbase * CC + kc, CC);
    tile_load_16x64(Bs, Wo + (size_t)nbase * CC + kc, CC);
    wait_tiles(0);
#pragma unroll
    for (int kk = 0; kk < 64; kk += 4) {
      v2f a = *(const v2f*)&As[lo][kk + 2 * hi];
      v2f bb = *(const v2f*)&Bs[lo][kk + 2 * hi];
      acc = wmma_f32(a, bb, acc);
    }
  }
  const float bval = bo[nbase + lo];
#pragma unroll
  for (int p = 0; p < 8; ++p) {
    out[(size_t)(mbase + p + 8 * hi) * CC + nbase + lo] = acc[p] + bval;
  }
}

// ---------------------------------------------------------------------------
extern "C" void kernel_launch(void* const* d_in, const int* in_sizes, int n_in,
                              void* d_out, int out_size, void* d_ws,
                              size_t ws_size, hipStream_t stream) {
  const float* audio = (const float*)d_in[0];
  const float* text = (const float*)d_in[1];
  const int* sl = (const int*)d_in[2];
  const int* tl = (const int*)d_in[3];
  const float* Wq = (const float*)d_in[4];
  const float* bq = (const float*)d_in[5];
  const float* Wk = (const float*)d_in[6];
  const float* bk = (const float*)d_in[7];
  const float* Wv = (const float*)d_in[8];
  const float* bv = (const float*)d_in[9];
  const float* Wo = (const float*)d_in[10];
  const float* bo = (const float*)d_in[11];
  float* out = (float*)d_out;

  // workspace: [3][B][H][L][DK] qkv (24 MB) + [B][L][C] ctx (8 MB)
  float* qkv = (float*)d_ws;
  float* ctx = qkv + (size_t)3 * BB * HH * LL * DKK;

  dim3 g1((BB * LL) / 16, CC / 16, 3);
  qkv_proj_kernel<<<g1, 32, 0, stream>>>(audio, text, Wq, bq, Wk, bk, Wv, bv,
                                         qkv);
  dim3 g2(LL / 16, HH, BB);
  attn_kernel<<<g2, 32, 0, stream>>>(qkv, sl, tl, ctx);
  dim3 g3((BB * LL) / 16, CC / 16, 1);
  oproj_kernel<<<g3, 32, 0, stream>>>(ctx, Wo, bo, out);
}